// GraphConvolution_23063974379819
// MI455X (gfx1250) — compile-verified
//
#include <hip/hip_runtime.h>

typedef __attribute__((ext_vector_type(2))) float v2f;
typedef __attribute__((ext_vector_type(8))) float v8f;
typedef int v4i_ __attribute__((vector_size(16)));   // int32x4, matches builtin param

#define GCN_N_NODES 100000
#define GCN_N_EDGES 1600000
#define GCN_IN_F    256
#define GCN_OUT_F   128
#define LDS_STRIDE  258   // 256 + 2 pad: shifts banks by 2 per row, keeps 16B groups intact

#define AS1 __attribute__((address_space(1)))
#define AS3 __attribute__((address_space(3)))

#if defined(__gfx1250__) && __has_builtin(__builtin_amdgcn_global_load_async_to_lds_b128) && \
    __has_builtin(__builtin_amdgcn_s_wait_asynccnt)
#define GCN_ASYNC_LDS 1
#endif

// ---------------------------------------------------------------------------
// Kernel 1: support[m,n] = sum_k x[m,k]*W[k,n] + b[n]  via V_WMMA_F32_16X16X4_F32
// grid.x = N_NODES/16 = 6250 blocks; block = 256 threads = 8 waves.
// Wave w computes the 16x16 tile (rows = 16*blockIdx.x.., cols = 16*w..).
// A tile (16x256 f32 = 16 KB) staged to LDS via async-to-LDS when available.
// ---------------------------------------------------------------------------
__launch_bounds__(256)
__global__ void gcn_gemm_bias_wmma(const float* __restrict__ x,
                                   const float* __restrict__ W,
                                   const float* __restrict__ bias,
                                   float* __restrict__ support) {
    __shared__ float ldsA[16 * LDS_STRIDE];

    const int m_base = blockIdx.x * 16;
    const int tid    = threadIdx.x;

    // Stage the 16x256 A tile: each thread moves one 16B chunk per pass,
    // 4 passes (256 threads x 16B = 4KB/pass). 16B chunks never cross a row.
    {
        const int row0 = tid >> 6;         // (tid*4) / 256
        const int col  = (tid & 63) * 4;
#pragma unroll
        for (int p = 0; p < 4; ++p) {
            const int row = row0 + p * 4;
            const float* gsrc = x + (size_t)(m_base + row) * GCN_IN_F + col;
            float* ldst = &ldsA[row * LDS_STRIDE + col];
#ifdef GCN_ASYNC_LDS
            __builtin_amdgcn_global_load_async_to_lds_b128(
                (AS1 v4i_*)(AS1 void*)gsrc,
                (AS3 v4i_*)(AS3 void*)ldst,
                /*offset=*/0, /*cpol=*/0);
#else
            *(float4*)ldst = *(const float4*)gsrc;
#endif
        }
    }
#ifdef GCN_ASYNC_LDS
    __builtin_amdgcn_s_wait_asynccnt(0);   // my wave's async copies landed in LDS
#endif
    __syncthreads();                       // everyone's copies visible

    const int wave = tid >> 5;        // 0..7 -> which 16-wide N tile
    const int lane = tid & 31;
    const int half = lane >> 4;       // 0: lanes 0-15, 1: lanes 16-31
    const int l16  = lane & 15;
    const int n    = wave * 16 + l16; // output column
    const int koff = half * 2;        // f32 A/B layout: half0 -> K 0,1; half1 -> K 2,3

    // C init = bias broadcast down the M dimension.
    v8f acc;
    const float bv = bias[n];
#pragma unroll
    for (int r = 0; r < 8; ++r) acc[r] = bv;

    const float* ldsRow = &ldsA[l16 * LDS_STRIDE];

#pragma unroll 4
    for (int kk = 0; kk < GCN_IN_F / 4; ++kk) {
        const int k4 = kk * 4;

        // A fragment: lane holds x[m_base + l16, k4+koff .. +1] (2 VGPRs)
        v2f a;
        a[0] = ldsRow[k4 + koff + 0];
        a[1] = ldsRow[k4 + koff + 1];

        // B fragment: lane holds W[k4+koff .. +1, n] (stride OUT_F between K rows)
        const float* wp = W + (k4 + koff) * GCN_OUT_F + n;
        v2f bf;
        bf[0] = wp[0];
        bf[1] = wp[GCN_OUT_F];

        acc = __builtin_amdgcn_wmma_f32_16x16x4_f32(false, a, false, bf,
                                                    (short)0, acc, false, false);
    }

    // C/D layout: VGPR r -> M = r (lanes 0-15) or M = 8+r (lanes 16-31), N = l16.
#pragma unroll
    for (int r = 0; r < 8; ++r) {
        const int row = m_base + r + half * 8;
        support[row * GCN_OUT_F + n] = acc[r];
    }
}

// ---------------------------------------------------------------------------
// Kernel 2: zero the accumulator (d_out). float4 stores, 4 elems/thread.
// ---------------------------------------------------------------------------
__global__ void gcn_zero(float4* __restrict__ out) {
    const int i = blockIdx.x * blockDim.x + threadIdx.x;
    out[i] = make_float4(0.f, 0.f, 0.f, 0.f);
}

// ---------------------------------------------------------------------------
// Kernel 3: out[row] += val * support[col] per edge, fp32 HW atomics.
// One wave per edge: edge scalars are wave-uniform (scalarized), each lane
// gathers 16B of support with one global_load_b128 and issues 4 f32 atomics.
// ---------------------------------------------------------------------------
__launch_bounds__(256)
__global__ void gcn_edge_scatter(const float* __restrict__ support,
                                 const int* __restrict__ erow,
                                 const int* __restrict__ ecol,
                                 const float* __restrict__ eval,
                                 float* __restrict__ out) {
    const int e    = blockIdx.x * 8 + (threadIdx.x >> 5);  // 8 waves/block, 1 edge/wave
    const int lane = threadIdx.x & 31;

    const int   r = erow[e];   // wave-uniform
    const int   c = ecol[e];
    const float v = eval[e];

    const float4 s = ((const float4*)(support + (size_t)c * GCN_OUT_F))[lane];
    float* o = out + (size_t)r * GCN_OUT_F + lane * 4;
    unsafeAtomicAdd(o + 0, v * s.x);
    unsafeAtomicAdd(o + 1, v * s.y);
    unsafeAtomicAdd(o + 2, v * s.z);
    unsafeAtomicAdd(o + 3, v * s.w);
}

// ---------------------------------------------------------------------------
// Kernel 4: in-place ReLU.
// ---------------------------------------------------------------------------
__global__ void gcn_relu(float4* __restrict__ out) {
    const int i = blockIdx.x * blockDim.x + threadIdx.x;
    float4 v = out[i];
    v.x = fmaxf(v.x, 0.f);
    v.y = fmaxf(v.y, 0.f);
    v.z = fmaxf(v.z, 0.f);
    v.w = fmaxf(v.w, 0.f);
    out[i] = v;
}

extern "C" void kernel_launch(void* const* d_in, const int* in_sizes, int n_in,
                              void* d_out, int out_size, void* d_ws, size_t ws_size,
                              hipStream_t stream) {
    const float* x    = (const float*)d_in[0];
    const int*   erow = (const int*)d_in[1];
    const int*   ecol = (const int*)d_in[2];
    const float* ev   = (const float*)d_in[3];
    const float* W    = (const float*)d_in[4];
    const float* b    = (const float*)d_in[5];
    float* out     = (float*)d_out;
    float* support = (float*)d_ws;   // 100000*128*4 = 51.2 MB scratch

    // 1) support = x @ W + b   (WMMA f32)
    gcn_gemm_bias_wmma<<<GCN_N_NODES / 16, 256, 0, stream>>>(x, W, b, support);

    // 2) out = 0  (fresh accumulator every call -> deterministic replay)
    const int nvec4 = GCN_N_NODES * GCN_OUT_F / 4;   // 3.2M float4
    gcn_zero<<<(nvec4 + 255) / 256, 256, 0, stream>>>((float4*)out);

    // 3) atomic scatter-add over edges (1 wave per edge)
    gcn_edge_scatter<<<GCN_N_EDGES / 8, 256, 0, stream>>>(support, erow, ecol, ev, out);

    // 4) relu in place
    gcn_relu<<<(nvec4 + 255) / 256, 256, 0, stream>>>((float4*)out);
}